// ginblock_70600672411873
// MI455X (gfx1250) — compile-verified
//
#include <hip/hip_runtime.h>

typedef __attribute__((ext_vector_type(2))) float v2f;
typedef __attribute__((ext_vector_type(8))) float v8f;

#define N_NODES 40000
#define N_EDGES 640000
#define D 128
#define LDSTRIDE 132   // 128 + 4 pad: lane L hits bank (4L+k)%64 -> conflict-free column reads

// ---------------------------------------------------------------- zero fill
__global__ void zero_f4(float4* __restrict__ p, int n4) {
  int i = blockIdx.x * blockDim.x + threadIdx.x;
  if (i < n4) p[i] = make_float4(0.f, 0.f, 0.f, 0.f);
}

// ---------------------------------------------------------------- edge scatter
// One wave32 per edge: 32 lanes x float4 = 128 floats. agg lives in d_out.
// Working set (ndata + agg + deg ~= 41 MB) is resident in the 192 MB L2,
// so the f32 atomics resolve in L2, not HBM.
__global__ void edge_scatter(const float* __restrict__ ndata,
                             const int* __restrict__ src,
                             const int* __restrict__ dst,
                             float* __restrict__ agg,
                             float* __restrict__ deg) {
  const int wave = threadIdx.x >> 5;
  const int lane = threadIdx.x & 31;
  const int e = blockIdx.x * 8 + wave;
  if (e >= N_EDGES) return;
  const int s = __builtin_amdgcn_readfirstlane(src[e]);
  const int d = __builtin_amdgcn_readfirstlane(dst[e]);
  const float4 v = *(const float4*)(ndata + (size_t)s * D + lane * 4);
  float* ap = agg + (size_t)d * D + lane * 4;
  atomicAdd(ap + 0, v.x);
  atomicAdd(ap + 1, v.y);
  atomicAdd(ap + 2, v.z);
  atomicAdd(ap + 3, v.w);
  if (lane == 0) atomicAdd(deg + d, 1.0f);
}

// ---------------------------------------------------------------- fused mean + GEMM
// Block = 256 threads = 8 waves. Block owns 16 output rows; wave w owns the
// 16x16 tile at columns w*16. h = ndata + agg/max(deg,1) staged in LDS, then
// K-loop of V_WMMA_F32_16X16X4_F32 (full f32 precision, matches reference).
__global__ void __launch_bounds__(256) gin_gemm(
    const float* __restrict__ ndata,
    const float* __restrict__ Wm,      // [128,128], out = h @ Wm^T + b
    const float* __restrict__ bias,
    const float* __restrict__ deg,
    float* __restrict__ out) {         // holds agg on entry, result on exit
  __shared__ float hs[16 * LDSTRIDE];

  const int row0 = blockIdx.x * 16;

  // ---- stage h tile (16 x 128) into LDS (consumes agg from `out`)
  for (int i = threadIdx.x; i < 16 * D; i += 256) {
    const int r = i >> 7;
    const int c = i & 127;
    const int row = row0 + r;
    const float rd = 1.0f / fmaxf(deg[row], 1.0f);
    hs[r * LDSTRIDE + c] =
        ndata[(size_t)row * D + c] + out[(size_t)row * D + c] * rd;
  }
  __syncthreads();

  // ---- WMMA compute: D = A(16x4) x B(4x16) + C, K = 128 in steps of 4
  const int wave    = threadIdx.x >> 5;
  const int lane    = threadIdx.x & 31;
  const int halfsel = lane >> 4;        // 0: lanes 0-15, 1: lanes 16-31
  const int mn      = lane & 15;        // M for A-frag, N for B/C/D-frags
  const int c0      = wave * 16;

  v8f acc = {0.f, 0.f, 0.f, 0.f, 0.f, 0.f, 0.f, 0.f};

#pragma unroll
  for (int k0 = 0; k0 < D; k0 += 4) {
    const int kb = k0 + 2 * halfsel;
    // A 16x4: lane(mn)=M, holds K = kb, kb+1  (LDS, conflict-free via pad)
    v2f a = *(const v2f*)(&hs[mn * LDSTRIDE + kb]);
    // B 4x16: lane(mn)=N, holds K = kb, kb+1; B[k][n] = W[n][k] (64 KB, L2-hot)
    v2f b = *(const v2f*)(&Wm[(size_t)(c0 + mn) * D + kb]);
    acc = __builtin_amdgcn_wmma_f32_16x16x4_f32(
        /*neg_a=*/false, a, /*neg_b=*/false, b,
        /*c_mod=*/(short)0, acc, /*reuse_a=*/false, /*reuse_b=*/false);
  }

  // ---- bias + store: VGPR r -> M = r + 8*halfsel, N = mn
  const float bv = bias[c0 + mn];
#pragma unroll
  for (int r = 0; r < 8; ++r) {
    const int m = r + 8 * halfsel;
    out[(size_t)(row0 + m) * D + c0 + mn] = acc[r] + bv;
  }
}

// ---------------------------------------------------------------- launch
extern "C" void kernel_launch(void* const* d_in, const int* in_sizes, int n_in,
                              void* d_out, int out_size, void* d_ws, size_t ws_size,
                              hipStream_t stream) {
  const float* ndata = (const float*)d_in[0];
  const int*   src   = (const int*)d_in[1];
  const int*   dst   = (const int*)d_in[2];
  const float* Wm    = (const float*)d_in[3];
  const float* bias  = (const float*)d_in[4];
  float* out = (float*)d_out;
  float* deg = (float*)d_ws;   // 40000 floats = 160 KB scratch

  // zero agg (aliased into d_out) and deg
  const int n4a = N_NODES * D / 4;   // 1,280,000 float4
  zero_f4<<<(n4a + 255) / 256, 256, 0, stream>>>((float4*)out, n4a);
  const int n4d = N_NODES / 4;       // 10,000 float4
  zero_f4<<<(n4d + 255) / 256, 256, 0, stream>>>((float4*)deg, n4d);

  // scatter-mean accumulation (L2-resident atomics)
  edge_scatter<<<N_EDGES / 8, 256, 0, stream>>>(ndata, src, dst, out, deg);

  // fused (ndata + mean) @ W^T + b via f32 WMMA
  gin_gemm<<<N_NODES / 16, 256, 0, stream>>>(ndata, Wm, bias, deg, out);
}